// StandardGeometryWithFBP_28802050687697
// MI455X (gfx1250) — compile-verified
//
#include <hip/hip_runtime.h>
#include <math.h>

// ---------------- geometry constants (match reference) ----------------
#define A_N 24
#define U_N 128
#define V_N 64
#define W_N 128
#define H_N 128
#define D_N 64
#define S_N 128

__device__ __constant__ const float k_SID  = 256.0f;
__device__ __constant__ const float k_SDD  = 512.0f;
#define HS_F   100.0f
#define STEP_F (2.0f * HS_F / (float)S_N)          // 1.5625 (integral weight)
#define ELL0_F (256.0f - HS_F)                     // 156
#define DELL_F (2.0f * HS_F / (float)(S_N - 1))    // linspace spacing
#define PI_F   3.14159265358979323846f

// ---------------- trilinear gather, zero outside (map_coordinates order=1, constant) ----
__device__ __forceinline__ float trilerp(const float* __restrict__ vol,
                                         float cz, float cy, float cx) {
    float zf = floorf(cz), yf = floorf(cy), xf = floorf(cx);
    int z0 = (int)zf, y0 = (int)yf, x0 = (int)xf;
    float fz = cz - zf, fy = cy - yf, fx = cx - xf;
    float acc = 0.0f;
#pragma unroll
    for (int dz = 0; dz < 2; ++dz) {
        int zi = z0 + dz;
        if ((unsigned)zi >= (unsigned)D_N) continue;
        float wz = dz ? fz : (1.0f - fz);
#pragma unroll
        for (int dy = 0; dy < 2; ++dy) {
            int yi = y0 + dy;
            if ((unsigned)yi >= (unsigned)H_N) continue;
            float wzy = wz * (dy ? fy : (1.0f - fy));
#pragma unroll
            for (int dx = 0; dx < 2; ++dx) {
                int xi = x0 + dx;
                if ((unsigned)xi >= (unsigned)W_N) continue;
                float w = wzy * (dx ? fx : (1.0f - fx));
                acc = fmaf(w, vol[(zi * H_N + yi) * W_N + xi], acc);
            }
        }
    }
    return acc;
}

// ---------------- shared ray setup ----------------
__device__ __forceinline__ void ray_setup(int a, int v, int u,
                                          float& sx, float& sy,
                                          float& ux, float& uy, float& uz) {
    float ang = (float)a * (2.0f * PI_F / (float)A_N);
    float s, c;
    __sincosf(ang, &s, &c);
    float uu = (float)u - (float)(U_N - 1) * 0.5f;   // u - 63.5
    float vv = (float)v - (float)(V_N - 1) * 0.5f;   // v - 31.5
    sx = -k_SID * c;
    sy = -k_SID * s;
    // dir = pix - src = (SDD*c - uu*s, SDD*s + uu*c, vv)
    float dx = k_SDD * c - uu * s;
    float dy = k_SDD * s + uu * c;
    float dz = vv;
    float rn = rsqrtf(dx * dx + dy * dy + dz * dz);
    ux = dx * rn; uy = dy * rn; uz = dz * rn;
}

// ---------------- K1: forward project + residual + cosine weight ----------------
__global__ void __launch_bounds__(256)
fbp_fp_residual_kernel(const float* __restrict__ vol,
                       const float* __restrict__ p,
                       float* __restrict__ res) {
    int idx = blockIdx.x * blockDim.x + threadIdx.x;
    if (idx >= A_N * V_N * U_N) return;
    int u = idx % U_N;
    int v = (idx / U_N) % V_N;
    int a = idx / (U_N * V_N);

    float sx, sy, ux, uy, uz;
    ray_setup(a, v, u, sx, sy, ux, uy, uz);

    float acc = 0.0f;
#pragma unroll 4
    for (int i = 0; i < S_N; ++i) {
        float ell = ELL0_F + (float)i * DELL_F;
        float px = fmaf(ell, ux, sx);
        float py = fmaf(ell, uy, sy);
        float pz = ell * uz;
        acc += trilerp(vol,
                       pz + (float)(D_N - 1) * 0.5f,
                       py + (float)(H_N - 1) * 0.5f,
                       px + (float)(W_N - 1) * 0.5f);
    }
    float sino = acc * STEP_F;
    // cosine weight: uses V/2 and U/2 (not (V-1)/2)
    float dv = (float)v - (float)V_N * 0.5f;
    float du = (float)u - (float)U_N * 0.5f;
    float cw = k_SDD * rsqrtf(k_SDD * k_SDD + dv * dv + du * du);
    res[idx] = (sino - p[idx]) * cw;
}

// ---------------- K2: build ramp-filter matrix F[k][n] = ramp(n-k+64) ----------------
__global__ void __launch_bounds__(256)
ramp_matrix_init_kernel(float* __restrict__ F) {
    int idx = blockIdx.x * blockDim.x + threadIdx.x;
    if (idx >= U_N * U_N) return;
    int n = idx % U_N;
    int k = idx / U_N;
    int d = n - k;                 // tap exists only for |d| <= 64
    float val = 0.0f;
    if (d == 0) {
        val = 0.125f;
    } else if ((d & 1) && (d >= -64) && (d <= 64)) {
        float fd = (float)d;
        val = -0.5f / (PI_F * PI_F * fd * fd);
    }
    F[idx] = val;
}

// ---------------- K3: ramp filter as f32 WMMA GEMM ----------------
// out[M=1536, N=128] = res[M,128] x F[128,128], one wave32 per 16x16 tile,
// V_WMMA_F32_16X16X4_F32 stepped over K by 4 (32 wmma per tile).
typedef float v2f __attribute__((ext_vector_type(2)));
typedef float v8f __attribute__((ext_vector_type(8)));

__global__ void __launch_bounds__(32)
ramp_wmma_kernel(const float* __restrict__ res,
                 const float* __restrict__ F,
                 float* __restrict__ out) {
    const int M_TILES_N = U_N / 16;            // 8 tiles along N
    int tile = blockIdx.x;                     // 0..767
    int tn = tile % M_TILES_N;
    int tm = tile / M_TILES_N;
    int lane = threadIdx.x & 31;
    int half = lane >> 4;                      // 0: lanes 0-15, 1: lanes 16-31
    int l = lane & 15;
    int m0 = tm * 16, n0 = tn * 16;

    v8f acc = {};
    for (int k = 0; k < U_N; k += 4) {
        // A 16x4 layout: lanes 0-15 hold (K=k,K=k+1) for row m0+l; lanes 16-31 hold (k+2,k+3)
        v2f afrag, bfrag;
        const float* arow = res + (m0 + l) * U_N + k + 2 * half;
        afrag.x = arow[0];
        afrag.y = arow[1];
        // B 4x16 layout (transpose of A layout): lanes 0-15 hold rows K=k,k+1 at col n0+l
        bfrag.x = F[(k + 2 * half) * U_N + n0 + l];
        bfrag.y = F[(k + 2 * half + 1) * U_N + n0 + l];
        acc = __builtin_amdgcn_wmma_f32_16x16x4_f32(
            /*neg_a=*/false, afrag, /*neg_b=*/false, bfrag,
            /*c_mod=*/(short)0, acc, /*reuse_a=*/false, /*reuse_b=*/false);
    }
    // C/D layout: VGPR r -> row m0 + r + 8*half, col n0 + l
#pragma unroll
    for (int r = 0; r < 8; ++r) {
        out[(m0 + r + 8 * half) * U_N + n0 + l] = acc[r];
    }
}

// ---------------- K4: exact adjoint (scatter) back projection ----------------
__global__ void __launch_bounds__(256)
fbp_bp_kernel(const float* __restrict__ filt,
              float* __restrict__ vol) {
    int idx = blockIdx.x * blockDim.x + threadIdx.x;
    if (idx >= A_N * V_N * U_N) return;
    int u = idx % U_N;
    int v = (idx / U_N) % V_N;
    int a = idx / (U_N * V_N);

    float sx, sy, ux, uy, uz;
    ray_setup(a, v, u, sx, sy, ux, uy, uz);

    float g = filt[idx] * STEP_F;   // LAMB == 1
    if (g == 0.0f) { /* still deterministic; fallthrough does no atomics with w*g==0 anyway */ }

    for (int i = 0; i < S_N; ++i) {
        float ell = ELL0_F + (float)i * DELL_F;
        float px = fmaf(ell, ux, sx);
        float py = fmaf(ell, uy, sy);
        float pz = ell * uz;
        float cz = pz + (float)(D_N - 1) * 0.5f;
        float cy = py + (float)(H_N - 1) * 0.5f;
        float cx = px + (float)(W_N - 1) * 0.5f;

        float zf = floorf(cz), yf = floorf(cy), xf = floorf(cx);
        int z0 = (int)zf, y0 = (int)yf, x0 = (int)xf;
        float fz = cz - zf, fy = cy - yf, fx = cx - xf;
#pragma unroll
        for (int dz = 0; dz < 2; ++dz) {
            int zi = z0 + dz;
            if ((unsigned)zi >= (unsigned)D_N) continue;
            float wz = dz ? fz : (1.0f - fz);
#pragma unroll
            for (int dy = 0; dy < 2; ++dy) {
                int yi = y0 + dy;
                if ((unsigned)yi >= (unsigned)H_N) continue;
                float wzy = wz * (dy ? fy : (1.0f - fy));
#pragma unroll
                for (int dx = 0; dx < 2; ++dx) {
                    int xi = x0 + dx;
                    if ((unsigned)xi >= (unsigned)W_N) continue;
                    float w = wzy * (dx ? fx : (1.0f - fx));
                    unsafeAtomicAdd(&vol[(zi * H_N + yi) * W_N + xi], w * g);
                }
            }
        }
    }
}

// ---------------- launcher ----------------
extern "C" void kernel_launch(void* const* d_in, const int* in_sizes, int n_in,
                              void* d_out, int out_size, void* d_ws, size_t ws_size,
                              hipStream_t stream) {
    const float* x = (const float*)d_in[0];   // [1,1,64,128,128]
    const float* p = (const float*)d_in[1];   // [1,1,24,64,128]
    float* out = (float*)d_out;               // [1,1,64,128,128]

    const int NRAY = A_N * V_N * U_N;         // 196608
    float* ws   = (float*)d_ws;
    float* res  = ws;                         // 196608 f32
    float* Fmat = ws + NRAY;                  // 16384 f32
    float* filt = ws + NRAY + U_N * U_N;      // 196608 f32

    // 1) forward project + residual + cosine weight
    fbp_fp_residual_kernel<<<(NRAY + 255) / 256, 256, 0, stream>>>(x, p, res);

    // 2) ramp matrix
    ramp_matrix_init_kernel<<<(U_N * U_N + 255) / 256, 256, 0, stream>>>(Fmat);

    // 3) ramp filter GEMM via V_WMMA_F32_16X16X4_F32 (one wave per 16x16 tile)
    int tiles = (A_N * V_N / 16) * (U_N / 16); // 96 * 8 = 768
    ramp_wmma_kernel<<<tiles, 32, 0, stream>>>(res, Fmat, filt);

    // 4) zero output volume (graph-capturable async memset)
    hipMemsetAsync(d_out, 0, (size_t)out_size * sizeof(float), stream);

    // 5) adjoint back projection (scatter with hardware f32 atomics)
    fbp_bp_kernel<<<(NRAY + 255) / 256, 256, 0, stream>>>(filt, out);
}